// PixCorr_91087666413632
// MI455X (gfx1250) — compile-verified
//
#include <hip/hip_runtime.h>
#include <math.h>

// ---------------------------------------------------------------------------
// PixCorr on MI455X (gfx1250, wave32).
//
// Roofline: 402 MB streamed once / 23.3 TB/s HBM  => ~17 us floor.
// ~0.5 GFLOP total => utterly bandwidth-bound; kernel is built around
// global_load_b128 streaming + cheap exact reductions.
//
// Cross-lane reduction uses V_WMMA_F32_16X16X4_F32 (exact f32 matmul):
//   A[m][k]: lane m (m<16) holds A[m][0]=p_m (VGPR0), A[m][1]=0 (VGPR1);
//            lane m+16 holds A[m][2]=p_{m+16}, A[m][3]=0.
//   B = all-ones 4x16  =>  D[m][n] = p_m + p_{m+16}  (broadcast over n).
//   C/D layout: lane l<16, VGPR r  -> D[r][l]     = p_r     + p_{r+16}
//               lane l>=16, VGPR r -> D[8+r][l-16] = p_{8+r} + p_{24+r}
//   Sum the 8 D VGPRs locally, then add the opposite half-wave's value
//   (ds_swizzle SWAPX16) -> every lane holds the exact 32-lane f32 sum.
// ---------------------------------------------------------------------------

typedef float v2f __attribute__((ext_vector_type(2)));
typedef float v8f __attribute__((ext_vector_type(8)));

__device__ __forceinline__ float wave_allreduce_f32(float p) {
    v2f a; a[0] = p;    a[1] = 0.0f;
    v2f b; b[0] = 1.0f; b[1] = 1.0f;
    v8f c = {};
    // (neg_a, A, neg_b, B, c_mod, C, reuse_a, reuse_b) -> v_wmma_f32_16x16x4_f32
    v8f d = __builtin_amdgcn_wmma_f32_16x16x4_f32(false, a, false, b,
                                                  (short)0, c, false, false);
    float t = ((d[0] + d[1]) + (d[2] + d[3])) + ((d[4] + d[5]) + (d[6] + d[7]));
    // swap half-waves (lane ^ 16): group-of-32 swizzle, xor=0x10, and=0x1f
    int o = __builtin_amdgcn_ds_swizzle(__float_as_int(t), 0x401F);
    return t + __int_as_float(o);
}

#define TPB   256   // 8 waves per block
#define SPLIT 4     // chunks per row -> 1024 blocks of parallelism

// Stage 1: one block per (row, chunk). Streams float4 (global_load_b128),
// accumulates 5 sums per thread, wave-level WMMA all-reduce, LDS combine,
// writes {sZ, sB, sZZ, sBB, sZB} per block to d_ws.
__global__ __launch_bounds__(TPB) void pixcorr_stage1(
    const float4* __restrict__ preds,   // B
    const float4* __restrict__ targs,   // Z
    float* __restrict__ stats,          // [gridDim.x][5]
    int D4, int chunk4)
{
    const int row   = blockIdx.x / SPLIT;
    const int chunk = blockIdx.x % SPLIT;
    const long base = (long)row * (long)D4 + (long)chunk * (long)chunk4;
    const float4* __restrict__ Zp = targs + base;
    const float4* __restrict__ Bp = preds + base;

    float sz = 0.f, sb = 0.f, szz = 0.f, sbb = 0.f, szb = 0.f;

    #pragma unroll 4
    for (int i = threadIdx.x; i < chunk4; i += TPB) {
        float4 z = Zp[i];
        float4 p = Bp[i];
        sz  += (z.x + z.y) + (z.z + z.w);
        sb  += (p.x + p.y) + (p.z + p.w);
        szz  = fmaf(z.x, z.x, fmaf(z.y, z.y, fmaf(z.z, z.z, fmaf(z.w, z.w, szz))));
        sbb  = fmaf(p.x, p.x, fmaf(p.y, p.y, fmaf(p.z, p.z, fmaf(p.w, p.w, sbb))));
        szb  = fmaf(z.x, p.x, fmaf(z.y, p.y, fmaf(z.z, p.z, fmaf(z.w, p.w, szb))));
    }

    // Wave-level exact all-reduce (all 32 lanes active -> EXEC all 1s for WMMA)
    sz  = wave_allreduce_f32(sz);
    sb  = wave_allreduce_f32(sb);
    szz = wave_allreduce_f32(szz);
    sbb = wave_allreduce_f32(sbb);
    szb = wave_allreduce_f32(szb);

    __shared__ float ls[TPB / 32][5];
    const int wave = threadIdx.x >> 5;
    if ((threadIdx.x & 31) == 0) {
        ls[wave][0] = sz;  ls[wave][1] = sb;
        ls[wave][2] = szz; ls[wave][3] = sbb; ls[wave][4] = szb;
    }
    __syncthreads();

    if (threadIdx.x == 0) {
        float t0 = 0.f, t1 = 0.f, t2 = 0.f, t3 = 0.f, t4 = 0.f;
        #pragma unroll
        for (int w = 0; w < TPB / 32; ++w) {
            t0 += ls[w][0]; t1 += ls[w][1]; t2 += ls[w][2];
            t3 += ls[w][3]; t4 += ls[w][4];
        }
        float* o = stats + (size_t)blockIdx.x * 5;
        o[0] = t0; o[1] = t1; o[2] = t2; o[3] = t3; o[4] = t4;
    }
}

// Stage 2: single block, 256 threads; thread n folds row n's SPLIT chunk
// stats (double precision scalar math), computes corr_n, then the block
// reduces 256 corr values (WMMA wave reduce + LDS) into the mean.
__global__ __launch_bounds__(256) void pixcorr_stage2(
    const float* __restrict__ stats,    // [N*SPLIT][5]
    float* __restrict__ out, int N, int D)
{
    const int n = threadIdx.x;          // N == 256 == blockDim.x
    double sz = 0.0, sb = 0.0, szz = 0.0, sbb = 0.0, szb = 0.0;
    #pragma unroll
    for (int c = 0; c < SPLIT; ++c) {
        const float* s = stats + ((size_t)n * SPLIT + c) * 5;
        sz += s[0]; sb += s[1]; szz += s[2]; sbb += s[3]; szb += s[4];
    }
    const double invD = 1.0 / (double)D;
    double num = szb - sz * sb * invD;
    double zn2 = szz - sz * sz * invD; if (zn2 < 0.0) zn2 = 0.0;
    double bn2 = sbb - sb * sb * invD; if (bn2 < 0.0) bn2 = 0.0;
    double den = sqrt(zn2) * sqrt(bn2) + 1e-6;   // EPS from reference
    float corr = (float)(num / den);

    // mean over N: no divergence before the WMMA reduce
    float t = wave_allreduce_f32(corr);

    __shared__ float ls[8];
    const int wave = threadIdx.x >> 5;
    if ((threadIdx.x & 31) == 0) ls[wave] = t;
    __syncthreads();

    if (threadIdx.x == 0) {
        float tot = 0.f;
        #pragma unroll
        for (int w = 0; w < 8; ++w) tot += ls[w];
        out[0] = tot / (float)N;
    }
}

extern "C" void kernel_launch(void* const* d_in, const int* in_sizes, int n_in,
                              void* d_out, int out_size, void* d_ws, size_t ws_size,
                              hipStream_t stream) {
    (void)n_in; (void)out_size; (void)ws_size;
    const float4* preds = (const float4*)d_in[0];   // B in the reference
    const float4* targs = (const float4*)d_in[1];   // Z in the reference
    float* out   = (float*)d_out;
    float* stats = (float*)d_ws;                    // 256*SPLIT*5 floats = 20 KB

    const int N  = 256;
    const int D  = in_sizes[0] / N;                 // 196608
    const int D4 = D / 4;                           // 49152 float4 per row
    const int chunk4 = D4 / SPLIT;                  // 12288 float4 per block

    pixcorr_stage1<<<N * SPLIT, TPB, 0, stream>>>(preds, targs, stats, D4, chunk4);
    pixcorr_stage2<<<1, 256, 0, stream>>>(stats, out, N, D);
}